// VectorQuantizer_40896678592816
// MI455X (gfx1250) — compile-verified
//
#include <hip/hip_runtime.h>
#include <hip/hip_bf16.h>

typedef __attribute__((ext_vector_type(2))) float v2f;
typedef __attribute__((ext_vector_type(8))) float v8f;

#define CODEBOOK_K 4096
#define DIM 128
#define BATCH 16
#define TLEN 2048
#define NTOT (BATCH * TLEN)   // 32768
#define LDS_STRIDE 132        // 128 + 4 pad: b64 reads hit all 64 banks once
#define NKTILES (CODEBOOK_K / 16)

// ---------------------------------------------------------------------------
// Kernel 1: enorm[k] = ||emb_k||^2.  One wave per code row.
// ---------------------------------------------------------------------------
__global__ __launch_bounds__(256) void vq_enorm_kernel(
    const float* __restrict__ emb, float* __restrict__ enorm)
{
    const int wave_id = (blockIdx.x * 256 + threadIdx.x) >> 5;  // code index
    const int lane    = threadIdx.x & 31;
    const float4 v = *(const float4*)(emb + (size_t)wave_id * DIM + lane * 4);
    float s = v.x * v.x + v.y * v.y + v.z * v.z + v.w * v.w;
#pragma unroll
    for (int off = 16; off > 0; off >>= 1)
        s += __shfl_xor(s, off, 32);
    if (lane == 0) enorm[wave_id] = s;
}

// ---------------------------------------------------------------------------
// Async copy: 16B global -> LDS, tracked by ASYNCcnt.  INST_OFFSET applies to
// both the LDS and memory addresses, so one address pair serves both halves
// of each thread's 32B slice.
// ---------------------------------------------------------------------------
__device__ __forceinline__ void async_b128(unsigned lds_off, const float* gsrc)
{
    asm volatile("global_load_async_to_lds_b128 %0, %1, off"
                 :: "v"(lds_off), "v"(gsrc) : "memory");
    asm volatile("global_load_async_to_lds_b128 %0, %1, off offset:16"
                 :: "v"(lds_off), "v"(gsrc) : "memory");
}

// ---------------------------------------------------------------------------
// Kernel 2: fused GEMM (WMMA f32 16x16x4) + running argmin.
// Block = 256 threads = 8 waves; each wave owns one 16-row z tile held in
// registers.  Block loops over 256 k-tiles of 16 codes; e-tiles are staged
// into double-buffered LDS with GLOBAL_LOAD_ASYNC_TO_LDS_B128, prefetched one
// tile ahead of the WMMA pipeline.
// ---------------------------------------------------------------------------
__global__ __launch_bounds__(256) void vq_argmin_wmma_kernel(
    const float* __restrict__ z, const float* __restrict__ emb,
    const float* __restrict__ enorm,
    float* __restrict__ idx_out_f, int* __restrict__ idx_out_i,
    float* __restrict__ d2min_out)
{
    __shared__ float lds_e[2][16 * LDS_STRIDE];

    const int tid   = threadIdx.x;
    const int wave  = tid >> 5;
    const int lane  = tid & 31;
    const int laneN = lane & 15;   // A-row index for loads / code column for C
    const int hi    = lane >> 4;   // 0: K={0,1}, 1: K={2,3} within each frag

    const int nbase = (blockIdx.x * 8 + wave) * 16;   // 16 consecutive n
    const int b     = nbase / TLEN;
    const int tbase = nbase % TLEN;                   // tile never crosses b

    // Per-thread staging slice: 8 floats of the 16x128 e-tile.
    const int srow = tid >> 4;          // 0..15
    const int scol = (tid & 15) * 8;    // 0..120

    // --- Load z tile into 32 A-fragments (16x4 f32 layout), registers only.
    // z layout is (B, D, T): element (n=b*T+t, d) at z[b*D*T + d*T + t].
    const float* zb = z + (size_t)b * DIM * TLEN + tbase;
    v2f afrag[32];
    float zsq = 0.f;
#pragma unroll
    for (int j = 0; j < 32; ++j) {
        const int d0 = 4 * j + 2 * hi;
        const float x = zb[(size_t)d0 * TLEN + laneN];
        const float y = zb[(size_t)(d0 + 1) * TLEN + laneN];
        afrag[j].x = x; afrag[j].y = y;
        zsq += x * x + y * y;
    }
    zsq += __shfl_xor(zsq, 16, 32);   // lanes L and L+16 hold same z row

    float minv[8];
    int   mini[8];
#pragma unroll
    for (int i = 0; i < 8; ++i) { minv[i] = 3.4e38f; mini[i] = 0; }

    // Prologue: async-stage tile 0 into buffer 0.
    async_b128((unsigned)(size_t)(&lds_e[0][srow * LDS_STRIDE + scol]),
               emb + (size_t)srow * DIM + scol);

    for (int kt = 0; kt < NKTILES; ++kt) {
        const int kbase = kt * 16;

        // Own async copies for tile kt done; barrier makes it workgroup-wide.
        asm volatile("s_wait_asynccnt 0x0" ::: "memory");
        __syncthreads();

        // Prefetch tile kt+1 into the other buffer (its last readers finished
        // compute kt-1 before the barrier above).
        if (kt + 1 < NKTILES) {
            async_b128((unsigned)(size_t)(&lds_e[(kt + 1) & 1]
                                                [srow * LDS_STRIDE + scol]),
                       emb + (size_t)(kbase + 16 + srow) * DIM + scol);
        }

        const float* cur = &lds_e[kt & 1][laneN * LDS_STRIDE + 2 * hi];

        // --- S = z_tile (16x128) @ e_tile^T (128x16): 32 WMMAs, two
        // independent accumulation chains.
        v8f acc0 = {0.f, 0.f, 0.f, 0.f, 0.f, 0.f, 0.f, 0.f};
        v8f acc1 = {0.f, 0.f, 0.f, 0.f, 0.f, 0.f, 0.f, 0.f};
#pragma unroll
        for (int j = 0; j < 32; j += 2) {
            // B 4x16 f32 layout: lane = N (code), v0 = K=0 (lo) / K=2 (hi),
            // v1 = K=1 / K=3  ->  a float2 from one emb row in LDS.
            v2f b0; b0.x = cur[4 * j];     b0.y = cur[4 * j + 1];
            v2f b1; b1.x = cur[4 * j + 4]; b1.y = cur[4 * j + 5];
            acc0 = __builtin_amdgcn_wmma_f32_16x16x4_f32(
                false, afrag[j],     false, b0, (short)0, acc0, false, false);
            acc1 = __builtin_amdgcn_wmma_f32_16x16x4_f32(
                false, afrag[j + 1], false, b1, (short)0, acc1, false, false);
        }

        // --- d2' = enorm[k] - 2 * S; update running argmin.
        const float en = enorm[kbase + laneN];
#pragma unroll
        for (int i = 0; i < 8; ++i) {
            const float val = en - 2.0f * (acc0[i] + acc1[i]);
            if (val < minv[i]) { minv[i] = val; mini[i] = kbase + laneN; }
        }
    }

    // --- Reduce (min, argmin) across the 16 code-columns in each half-wave.
#pragma unroll
    for (int off = 1; off < 16; off <<= 1) {
#pragma unroll
        for (int i = 0; i < 8; ++i) {
            const float ov = __shfl_xor(minv[i], off, 32);
            const int   oi = __shfl_xor(mini[i], off, 32);
            if (ov < minv[i] || (ov == minv[i] && oi < mini[i])) {
                minv[i] = ov; mini[i] = oi;
            }
        }
    }

    // znorm for row (8*hi + i): stored at lane (8*hi + i) after the xor-16 add.
    float zrow[8];
#pragma unroll
    for (int i = 0; i < 8; ++i) zrow[i] = __shfl(zsq, 8 * hi + i, 32);

    if (laneN == 0) {
#pragma unroll
        for (int i = 0; i < 8; ++i) {
            const int n = nbase + 8 * hi + i;   // VGPR i <-> row i (+8 hi half)
            idx_out_i[n] = mini[i];
            idx_out_f[n] = (float)mini[i];
            d2min_out[n] = zrow[i] + minv[i];   // == ||z_n - e_idx||^2
        }
    }
}

// ---------------------------------------------------------------------------
// Kernel 3: gather + straight-through write, replicating fl(z + fl(q - z)).
// ---------------------------------------------------------------------------
__global__ __launch_bounds__(256) void vq_quantize_kernel(
    const float* __restrict__ z, const float* __restrict__ emb,
    const int* __restrict__ idx, float* __restrict__ out)
{
    const size_t g = (size_t)blockIdx.x * 256 + threadIdx.x;  // over B*D*T
    const int t = (int)(g % TLEN);
    const size_t r = g / TLEN;
    const int d = (int)(r % DIM);
    const int b = (int)(r / DIM);
    const int k = idx[b * TLEN + t];
    const float zz = z[g];
    const float q  = emb[(size_t)k * DIM + d];
    out[g] = zz + (q - zz);
}

// ---------------------------------------------------------------------------
// Kernel 4: commit_loss = sum(d2min) / (N * D).  Single block.
// ---------------------------------------------------------------------------
__global__ __launch_bounds__(256) void vq_loss_kernel(
    const float* __restrict__ d2min, float* __restrict__ out)
{
    __shared__ float sm[256];
    float s = 0.f;
    for (int i = threadIdx.x; i < NTOT; i += 256) s += d2min[i];
    sm[threadIdx.x] = s;
    __syncthreads();
#pragma unroll
    for (int o = 128; o > 0; o >>= 1) {
        if (threadIdx.x < o) sm[threadIdx.x] += sm[threadIdx.x + o];
        __syncthreads();
    }
    if (threadIdx.x == 0) out[0] = sm[0] / (float)((size_t)NTOT * DIM);
}

// ---------------------------------------------------------------------------
extern "C" void kernel_launch(void* const* d_in, const int* in_sizes, int n_in,
                              void* d_out, int out_size, void* d_ws, size_t ws_size,
                              hipStream_t stream)
{
    (void)in_sizes; (void)n_in; (void)out_size; (void)ws_size;
    const float* z   = (const float*)d_in[0];   // (B, D, T) f32
    const float* emb = (const float*)d_in[1];   // (K, D) f32

    float* out   = (float*)d_out;
    float* quant = out;                                   // B*D*T floats
    float* idxf  = out + (size_t)BATCH * DIM * TLEN;      // N floats
    float* loss  = idxf + NTOT;                           // 1 float

    float* enorm = (float*)d_ws;                          // K floats
    float* d2min = enorm + CODEBOOK_K;                    // N floats
    int*   idxi  = (int*)(d2min + NTOT);                  // N ints

    vq_enorm_kernel<<<CODEBOOK_K / 8, 256, 0, stream>>>(emb, enorm);
    vq_argmin_wmma_kernel<<<NTOT / 128, 256, 0, stream>>>(
        z, emb, enorm, idxf, idxi, d2min);
    vq_quantize_kernel<<<(BATCH * DIM * TLEN) / 256, 256, 0, stream>>>(
        z, emb, idxi, quant);
    vq_loss_kernel<<<1, 256, 0, stream>>>(d2min, loss);
}